// AttnDecoderRNN_77549929497285
// MI455X (gfx1250) — compile-verified
//
#include <hip/hip_runtime.h>
#include <math.h>

typedef __attribute__((ext_vector_type(2))) float v2f;
typedef __attribute__((ext_vector_type(8))) float v8f;

#define HID 1024
#define VOC 50257
#define MAXLEN 150

// ---------------------------------------------------------------------------
// Generic fp32 GEMV  y = W x + b  (W row-major [rows x cols], cols % 4 == 0)
// One wave per 16-row tile, driven by V_WMMA_F32_16X16X4_F32.
// A tile: lane L holds row (L&15), K columns (L>>4)*2 + {0,1}  (ISA 7.12.2)
// B: x chunk in matrix column 0 only (lanes 0 and 16), other lanes zeroed.
// D column 0: lane 0 -> rows m+0..7, lane 16 -> rows m+8..15.
// ---------------------------------------------------------------------------
__global__ __launch_bounds__(256)
void gemv_wmma_f32(const float* __restrict__ W, const float* __restrict__ x,
                   const float* __restrict__ bias, float* __restrict__ y,
                   int rows, int cols) {
  const int lane = threadIdx.x & 31;
  const int tile = blockIdx.x * (blockDim.x >> 5) + (threadIdx.x >> 5);
  const int row0 = tile << 4;
  if (row0 >= rows) return;                       // wave-uniform exit
  const int m  = lane & 15;
  const int kh = lane >> 4;                       // K-half: 0 or 1
  int r = row0 + m; if (r >= rows) r = rows - 1;  // clamp OOB rows (dup reads)
  const float* __restrict__ Wp = W + (size_t)r * (size_t)cols + (kh << 1);
  const float* __restrict__ xp = x + (kh << 1);
  const float msk = (m == 0) ? 1.0f : 0.0f;       // only lanes 0 & 16 carry B
  v8f acc = {};
  #pragma unroll 4
  for (int kb = 0; kb < cols; kb += 4) {
    v2f a  = *(const v2f*)(Wp + kb);              // A[m][2kh], A[m][2kh+1]
    v2f bx = *(const v2f*)(xp + kb);              // x[kb+2kh], x[kb+2kh+1]
    bx.x *= msk; bx.y *= msk;                     // zero cols 1..15 of B
    acc = __builtin_amdgcn_wmma_f32_16x16x4_f32(
        /*neg_a=*/false, a, /*neg_b=*/false, bx,
        /*c_mod=*/(short)0, acc, /*reuse_a=*/false, /*reuse_b=*/false);
  }
  if (m == 0) {                                   // lanes 0 and 16 own col N=0
    const int base = row0 + (kh << 3);
    #pragma unroll
    for (int i = 0; i < 8; ++i) {
      int rr = base + i;
      if (rr < rows) y[rr] = acc[i] + bias[rr];
    }
  }
}

// Build attn_in = [emb_row ; h0]; comb_in[0:H] = emb_row.
__global__ void k_prep(const int* __restrict__ idx,
                       const float* __restrict__ hidden,
                       const float* __restrict__ emb_W,
                       float* __restrict__ attn_in,
                       float* __restrict__ comb_in) {
  int i = blockIdx.x * blockDim.x + threadIdx.x;
  if (i < HID) {
    float e = emb_W[(size_t)idx[0] * HID + i];
    attn_in[i]       = e;
    comb_in[i]       = e;
    attn_in[HID + i] = hidden[i];
  }
}

// softmax over 150 scores (in-place) + copy weights to d_out tail
__global__ __launch_bounds__(256)
void k_softmax_attn(float* __restrict__ scores, float* __restrict__ out_attn) {
  __shared__ float red[256];
  const int t = threadIdx.x;
  float v = (t < MAXLEN) ? scores[t] : -3.4e38f;
  red[t] = v; __syncthreads();
  for (int s = 128; s > 0; s >>= 1) {
    if (t < s) red[t] = fmaxf(red[t], red[t + s]);
    __syncthreads();
  }
  const float mx = red[0]; __syncthreads();
  float e = (t < MAXLEN) ? expf(v - mx) : 0.0f;
  red[t] = e; __syncthreads();
  for (int s = 128; s > 0; s >>= 1) {
    if (t < s) red[t] += red[t + s];
    __syncthreads();
  }
  const float inv = 1.0f / red[0];
  if (t < MAXLEN) {
    float w = e * inv;
    scores[t]   = w;
    out_attn[t] = w;
  }
}

// comb_in[H + h] = sum_j w[j] * enc[j][h]   (coalesced column reduction)
__global__ void k_attn_apply(const float* __restrict__ wgt,
                             const float* __restrict__ enc,
                             float* __restrict__ comb_in_hi) {
  int h = blockIdx.x * blockDim.x + threadIdx.x;
  if (h < HID) {
    float s = 0.0f;
    #pragma unroll 5
    for (int j = 0; j < MAXLEN; ++j) s = fmaf(wgt[j], enc[(size_t)j * HID + h], s);
    comb_in_hi[h] = s;
  }
}

__global__ void k_relu(const float* __restrict__ in, float* __restrict__ out, int n) {
  int i = blockIdx.x * blockDim.x + threadIdx.x;
  if (i < n) out[i] = fmaxf(in[i], 0.0f);
}

// GRU gate combine (PyTorch order r,z,n); writes h_new to ws and d_out
__global__ void k_gru_gates(const float* __restrict__ gi, const float* __restrict__ gh,
                            const float* __restrict__ h0,
                            float* __restrict__ h_ws, float* __restrict__ h_out) {
  int i = blockIdx.x * blockDim.x + threadIdx.x;
  if (i < HID) {
    float r = 1.0f / (1.0f + expf(-(gi[i]           + gh[i])));
    float z = 1.0f / (1.0f + expf(-(gi[HID + i]     + gh[HID + i])));
    float n = tanhf(gi[2 * HID + i] + r * gh[2 * HID + i]);
    float h = (1.0f - z) * n + z * h0[i];
    h_ws[i]  = h;
    h_out[i] = h;
  }
}

// log_softmax over VOC logits (single 1024-thread block; data stays in L2)
__global__ __launch_bounds__(1024)
void k_log_softmax(const float* __restrict__ logits, float* __restrict__ out) {
  __shared__ float red[1024];
  const int t = threadIdx.x;
  float mx = -3.4e38f;
  for (int i = t; i < VOC; i += 1024) mx = fmaxf(mx, logits[i]);
  red[t] = mx; __syncthreads();
  for (int s = 512; s > 0; s >>= 1) {
    if (t < s) red[t] = fmaxf(red[t], red[t + s]);
    __syncthreads();
  }
  mx = red[0]; __syncthreads();
  float sm = 0.0f;
  for (int i = t; i < VOC; i += 1024) sm += expf(logits[i] - mx);
  red[t] = sm; __syncthreads();
  for (int s = 512; s > 0; s >>= 1) {
    if (t < s) red[t] += red[t + s];
    __syncthreads();
  }
  const float lse = mx + logf(red[0]);
  for (int i = t; i < VOC; i += 1024) out[i] = logits[i] - lse;
}

static inline int gemv_blocks(int rows) {
  int tiles = (rows + 15) / 16;
  return (tiles + 7) / 8;           // 8 waves (16-row tiles) per 256-thread block
}

extern "C" void kernel_launch(void* const* d_in, const int* in_sizes, int n_in,
                              void* d_out, int out_size, void* d_ws, size_t ws_size,
                              hipStream_t stream) {
  const int*   idx    = (const int*)  d_in[0];
  const float* hidden = (const float*)d_in[1];
  const float* enc    = (const float*)d_in[2];
  const float* emb_W  = (const float*)d_in[3];
  const float* attn_W = (const float*)d_in[4];
  const float* attn_b = (const float*)d_in[5];
  const float* comb_W = (const float*)d_in[6];
  const float* comb_b = (const float*)d_in[7];
  const float* w_ih   = (const float*)d_in[8];
  const float* w_hh   = (const float*)d_in[9];
  const float* b_ih   = (const float*)d_in[10];
  const float* b_hh   = (const float*)d_in[11];
  const float* out_W  = (const float*)d_in[12];
  const float* out_b  = (const float*)d_in[13];

  float* out = (float*)d_out;             // [VOC logp | HID h_new | MAXLEN attn_w]
  float* ws  = (float*)d_ws;
  float* attn_in  = ws;                   // 2048
  float* comb_in  = ws + 2048;            // 2048
  float* scores   = ws + 4096;            // 256 (150 used)
  float* comb_out = ws + 4352;            // 1024
  float* xrelu    = ws + 5376;            // 1024
  float* gi       = ws + 6400;            // 3072
  float* gh       = ws + 9472;            // 3072
  float* hnew     = ws + 12544;           // 1024
  float* logits   = ws + 13568;           // 50257

  // 1) embedding gather + concat inputs
  k_prep<<<4, 256, 0, stream>>>(idx, hidden, emb_W, attn_in, comb_in);
  // 2) attention scores: [150 x 2048] @ attn_in + attn_b
  gemv_wmma_f32<<<gemv_blocks(MAXLEN), 256, 0, stream>>>(attn_W, attn_in, attn_b,
                                                         scores, MAXLEN, 2 * HID);
  // 3) softmax over 150 (also emits attn_weights output)
  k_softmax_attn<<<1, 256, 0, stream>>>(scores, out + VOC + HID);
  // 4) context vector -> comb_in[H:2H]
  k_attn_apply<<<4, 256, 0, stream>>>(scores, enc, comb_in + HID);
  // 5) combine: [1024 x 2048] @ comb_in + comb_b, then ReLU
  gemv_wmma_f32<<<gemv_blocks(HID), 256, 0, stream>>>(comb_W, comb_in, comb_b,
                                                      comb_out, HID, 2 * HID);
  k_relu<<<4, 256, 0, stream>>>(comb_out, xrelu, HID);
  // 6) GRU matvecs: [3072 x 1024] each
  gemv_wmma_f32<<<gemv_blocks(3 * HID), 256, 0, stream>>>(w_ih, xrelu, b_ih,
                                                          gi, 3 * HID, HID);
  gemv_wmma_f32<<<gemv_blocks(3 * HID), 256, 0, stream>>>(w_hh, hidden, b_hh,
                                                          gh, 3 * HID, HID);
  // 7) gate combine -> h_new (ws + d_out)
  k_gru_gates<<<4, 256, 0, stream>>>(gi, gh, hidden, hnew, out + VOC);
  // 8) logits: [50257 x 1024] @ h_new + out_b  (dominant, ~206 MB stream)
  gemv_wmma_f32<<<gemv_blocks(VOC), 256, 0, stream>>>(out_W, hnew, out_b,
                                                      logits, VOC, HID);
  // 9) log_softmax -> d_out[0:VOC]
  k_log_softmax<<<1, 1024, 0, stream>>>(logits, out);
}